// GlobalJointAttention_75479755260053
// MI455X (gfx1250) — compile-verified
//
#include <hip/hip_runtime.h>
#include <math.h>

// ---------------------------------------------------------------------------
// Problem constants (from reference: B=1, HW=576, S=4, Nt=577, C=768)
// ---------------------------------------------------------------------------
constexpr int kC    = 768;
constexpr int kC3   = 2304;                 // 3*C
constexpr int kHW   = 576;                  // image tokens
constexpr int kSTpl = 4;                    // templates
constexpr int kNtT  = 577;                  // tokens per template
constexpr int kN    = kHW + kSTpl * kNtT;   // 2884 total tokens
constexpr int kNH   = 12;                   // heads
constexpr int kHD   = 64;                   // head dim
constexpr int kMT   = (kN + 15) / 16;       // 181 row tiles

typedef float v2f __attribute__((ext_vector_type(2)));
typedef float v8f __attribute__((ext_vector_type(8)));

__device__ __forceinline__ v8f wmma_f32(v2f a, v2f b, v8f c) {
  // D = A(16x4,f32) * B(4x16,f32) + C(16x16,f32)
  return __builtin_amdgcn_wmma_f32_16x16x4_f32(
      /*neg_a=*/false, a, /*neg_b=*/false, b,
      /*c_mod=*/(short)0, c, /*reuse_a=*/false, /*reuse_b=*/false);
}

__device__ __forceinline__ v8f v8f_zero() {
  v8f z = {0.f, 0.f, 0.f, 0.f, 0.f, 0.f, 0.f, 0.f};
  return z;
}

// ---------------------------------------------------------------------------
// Async global->LDS staging (CDNA5 GLOBAL_LOAD_ASYNC_TO_LDS_B128, ASYNCcnt),
// with a safe synchronous fallback if the builtin is unavailable.
// Builtin signature (from hipcc diagnostic): (v4i AS1*, v4i AS3*, Ii, Ii).
// ---------------------------------------------------------------------------
#if defined(__has_builtin)
#if __has_builtin(__builtin_amdgcn_global_load_async_to_lds_b128)
#define USE_ASYNC_LDS 1
#endif
#endif
#ifndef USE_ASYNC_LDS
#define USE_ASYNC_LDS 0
#endif

typedef int v4i __attribute__((vector_size(16)));
typedef __attribute__((address_space(1))) v4i as1_v4i;
typedef __attribute__((address_space(3))) v4i as3_v4i;

__device__ __forceinline__ void copy16_to_lds(const float* g, float* l) {
#if USE_ASYNC_LDS
  __builtin_amdgcn_global_load_async_to_lds_b128((as1_v4i*)g, (as3_v4i*)l, 0, 0);
#else
  *(float4*)l = *(const float4*)g;
#endif
}

#if USE_ASYNC_LDS
#define WAIT_ASYNC(n) asm volatile("s_wait_asynccnt " #n ::: "memory")
#else
#define WAIT_ASYNC(n)
#endif

// ---------------------------------------------------------------------------
// Kernel 1: QKV GEMM.  X (kN x kC, gathered from q_tokens|t_tokens) @ W_qkv
// (kC x kC3) + b_qkv -> qkv (kN x kC3).
// Block = 8 waves sharing one 64-wide column strip; each wave owns a 16-row
// tile.  W k-slabs (16x64) are double-buffer async-staged into LDS and shared
// by all 8 waves (8x cut on W traffic).
// ---------------------------------------------------------------------------
__global__ __launch_bounds__(256) void qkv_gemm_kernel(
    const float* __restrict__ qtok, const float* __restrict__ ttok,
    const float* __restrict__ W, const float* __restrict__ bias,
    float* __restrict__ out) {
  __shared__ float sW[2][16 * 64];
  int wid = threadIdx.x >> 5;
  int lane = threadIdx.x & 31;
  int lr = lane & 15;
  int lh = lane >> 4;
  int nt = blockIdx.y;                       // 64-col strip
  int mt = blockIdx.x * 8 + wid;             // 16-row tile
  if (mt >= kMT) mt = kMT - 1;               // keep wave alive for barriers

  int row = mt * 16 + lr;
  int rowc = row < kN ? row : kN - 1;
  const float* xrow = (rowc < kHW) ? (qtok + (size_t)rowc * kC)
                                   : (ttok + (size_t)(rowc - kHW) * kC);

  // staging slice: thread t covers W[chunk*16 + t/16][nt*64 + (t%16)*4 .. +3]
  int srow = threadIdx.x >> 4;
  int scol = (threadIdx.x & 15) * 4;
  const float* wsrc = W + (size_t)srow * kC3 + nt * 64 + scol;

  v8f acc[4];
#pragma unroll
  for (int c = 0; c < 4; ++c) acc[c] = v8f_zero();

  constexpr int NC = kC / 16;                // 48 k-chunks
  copy16_to_lds(wsrc, &sW[0][srow * 64 + scol]);
  for (int c = 0; c < NC; ++c) {
    int cur = c & 1;
    if (c + 1 < NC) {
      copy16_to_lds(wsrc + (size_t)(c + 1) * 16 * kC3,
                    &sW[cur ^ 1][srow * 64 + scol]);
      WAIT_ASYNC(1);
    } else {
      WAIT_ASYNC(0);
    }
    __syncthreads();
    const float* wl = sW[cur];
#pragma unroll
    for (int ks = 0; ks < 4; ++ks) {
      int d0 = ks * 4 + 2 * lh;
      v2f a = *(const v2f*)(xrow + c * 16 + d0);
#pragma unroll
      for (int cg = 0; cg < 4; ++cg) {
        v2f b;
        b.x = wl[d0 * 64 + cg * 16 + lr];
        b.y = wl[(d0 + 1) * 64 + cg * 16 + lr];
        acc[cg] = wmma_f32(a, b, acc[cg]);
      }
    }
    __syncthreads();
  }

#pragma unroll
  for (int cg = 0; cg < 4; ++cg) {
    int col = nt * 64 + cg * 16 + lr;
    float bv = bias[col];
#pragma unroll
    for (int r = 0; r < 8; ++r) {
      int orow = mt * 16 + r + 8 * lh;
      if (orow < kN) out[(size_t)orow * kC3 + col] = acc[cg][r] + bv;
    }
  }
}

// ---------------------------------------------------------------------------
// Kernel 2: RoPE2D (image tokens) + SO3-PRoPE (template image tokens),
// in place on Q and K slices of the qkv buffer.
// ---------------------------------------------------------------------------
__global__ __launch_bounds__(256) void rope_so3_kernel(
    float* __restrict__ qkv, const float* __restrict__ pos2d,
    const float* __restrict__ poses) {
  int idx = blockIdx.x * blockDim.x + threadIdx.x;
  if (idx >= 2 * kN * kNH) return;
  int which = idx / (kN * kNH);  // 0 = q, 1 = k
  int rem = idx % (kN * kNH);
  int n = rem / kNH;
  int h = rem % kNH;
  float* t = qkv + (size_t)n * kC3 + which * kC + h * kHD;

  if (n < kHW) {
    float py = pos2d[n * 2 + 0];
    float px = pos2d[n * 2 + 1];
#pragma unroll
    for (int i = 0; i < 16; ++i) {
      float invf = powf(100.0f, -(float)i * (1.0f / 16.0f));
      float ay = py * invf;
      float cy = cosf(ay), sy = sinf(ay);
      float a0 = t[i], b0 = t[i + 16];
      t[i] = a0 * cy - b0 * sy;
      t[i + 16] = b0 * cy + a0 * sy;
      float ax = px * invf;
      float cx = cosf(ax), sx = sinf(ax);
      float a1 = t[32 + i], b1 = t[48 + i];
      t[32 + i] = a1 * cx - b1 * sx;
      t[48 + i] = b1 * cx + a1 * sx;
    }
  } else {
    int tn = n - kHW;
    int sIdx = tn / kNtT;
    int ti = tn % kNtT;
    if (ti < kHW) {  // "image" part of template tokens (num_img = HW)
      const float* R = poses + (size_t)sIdx * 16;  // 4x4 row-major
      float r00 = R[0], r01 = R[1], r02 = R[2];
      float r10 = R[4], r11 = R[5], r12 = R[6];
      float r20 = R[8], r21 = R[9], r22 = R[10];
#pragma unroll
      for (int kv = 0; kv < 10; ++kv) {
        float v0 = t[3 * kv], v1 = t[3 * kv + 1], v2 = t[3 * kv + 2];
        t[3 * kv]     = r00 * v0 + r01 * v1 + r02 * v2;
        t[3 * kv + 1] = r10 * v0 + r11 * v1 + r12 * v2;
        t[3 * kv + 2] = r20 * v0 + r21 * v1 + r22 * v2;
      }
    }
  }
}

// ---------------------------------------------------------------------------
// Kernel 3: flash attention.  Block = 8 waves, one head, 8 consecutive
// 16-query tiles.  K/V tiles double-buffer async-staged into LDS and shared
// across the 8 waves; online softmax per wave; S and P*V via f32 WMMA.
// ---------------------------------------------------------------------------
__global__ __launch_bounds__(256) void attn_kernel(
    const float* __restrict__ qkv, float* __restrict__ attn_out) {
  __shared__ float sK[2][16 * 64];
  __shared__ float sV[2][16 * 64];
  __shared__ float sP[8][256];

  int wid = threadIdx.x >> 5;
  int lane = threadIdx.x & 31;
  int lr = lane & 15;
  int lh = lane >> 4;
  int h = blockIdx.y;
  int qt = blockIdx.x * 8 + wid;
  if (qt >= kMT) qt = kMT - 1;  // keep wave alive for barriers (dup stores OK)

  const float scale = 0.125f;  // HEAD_DIM^-0.5

  int qrow = qt * 16 + lr;
  int qrowc = qrow < kN ? qrow : kN - 1;
  const float* qptr = qkv + (size_t)qrowc * kC3 + h * kHD;
  v2f qa[16];
#pragma unroll
  for (int kk = 0; kk < 16; ++kk) {
    int d0 = kk * 4 + 2 * lh;
    v2f q = *(const v2f*)(qptr + d0);
    qa[kk].x = q.x * scale;  // pre-scale Q
    qa[kk].y = q.y * scale;
  }
  const float* kbase = qkv + kC + h * kHD;
  const float* vbase = qkv + 2 * kC + h * kHD;

  // staging slice: thread t covers row t/16, cols (t%16)*4..+3 of the tile
  int srow = threadIdx.x >> 4;
  int scol = (threadIdx.x & 15) * 4;

  float m[8], ssum[8];
  v8f o[4];
#pragma unroll
  for (int r = 0; r < 8; ++r) { m[r] = -INFINITY; ssum[r] = 0.0f; }
#pragma unroll
  for (int c = 0; c < 4; ++c) o[c] = v8f_zero();

  float* myP = sP[wid];

  // prologue: stage tile 0
  {
    int gr = srow;  // tile 0 rows 0..15 always < kN
    copy16_to_lds(kbase + (size_t)gr * kC3 + scol, &sK[0][srow * 64 + scol]);
    copy16_to_lds(vbase + (size_t)gr * kC3 + scol, &sV[0][srow * 64 + scol]);
  }

  for (int j = 0; j < kMT; ++j) {
    int cur = j & 1;
    if (j + 1 < kMT) {
      int gr = (j + 1) * 16 + srow;
      if (gr >= kN) gr = kN - 1;  // clamped rows are softmax-masked / P==0
      copy16_to_lds(kbase + (size_t)gr * kC3 + scol,
                    &sK[cur ^ 1][srow * 64 + scol]);
      copy16_to_lds(vbase + (size_t)gr * kC3 + scol,
                    &sV[cur ^ 1][srow * 64 + scol]);
      WAIT_ASYNC(2);
    } else {
      WAIT_ASYNC(0);
    }
    __syncthreads();

    const float* kl = sK[cur];
    const float* vl = sV[cur];
    int key0 = j * 16;

    // --- S = Q K^T (16 WMMA) ---
    v8f s = v8f_zero();
#pragma unroll
    for (int kk = 0; kk < 16; ++kk) {
      int d0 = kk * 4 + 2 * lh;
      v2f b = *(const v2f*)(kl + lr * 64 + d0);
      s = wmma_f32(qa[kk], b, s);
    }

    // --- online softmax ---
    bool oob = (key0 + lr) >= kN;
    float alpha[8];
#pragma unroll
    for (int r = 0; r < 8; ++r) {
      float sv = oob ? -INFINITY : s[r];
      float rm = sv;
      rm = fmaxf(rm, __shfl_xor(rm, 1, 32));
      rm = fmaxf(rm, __shfl_xor(rm, 2, 32));
      rm = fmaxf(rm, __shfl_xor(rm, 4, 32));
      rm = fmaxf(rm, __shfl_xor(rm, 8, 32));
      float nm = fmaxf(m[r], rm);             // finite: each tile has valid keys
      alpha[r] = __expf(m[r] - nm);           // first iter: exp(-inf) = 0
      m[r] = nm;
      float p = __expf(sv - nm);              // masked lanes -> exactly 0
      s[r] = p;
      float ps = p;
      ps += __shfl_xor(ps, 1, 32);
      ps += __shfl_xor(ps, 2, 32);
      ps += __shfl_xor(ps, 4, 32);
      ps += __shfl_xor(ps, 8, 32);
      ssum[r] = ssum[r] * alpha[r] + ps;
    }
#pragma unroll
    for (int r = 0; r < 8; ++r) {
      o[0][r] *= alpha[r];
      o[1][r] *= alpha[r];
      o[2][r] *= alpha[r];
      o[3][r] *= alpha[r];
    }

    // --- C-layout -> A-fragment layout for P via private LDS slab ---
#pragma unroll
    for (int r = 0; r < 8; ++r) myP[(r + 8 * lh) * 16 + lr] = s[r];
    asm volatile("s_wait_dscnt 0x0" ::: "memory");
    v2f pa[4];
#pragma unroll
    for (int kk = 0; kk < 4; ++kk) {
      int d0 = kk * 4 + 2 * lh;
      pa[kk] = *(const v2f*)(myP + lr * 16 + d0);
    }

    // --- O += P * V (16 WMMA) ---
#pragma unroll
    for (int kk = 0; kk < 4; ++kk) {
      int vr = kk * 4 + 2 * lh;
#pragma unroll
      for (int cg = 0; cg < 4; ++cg) {
        v2f b;
        b.x = vl[vr * 64 + cg * 16 + lr];
        b.y = vl[(vr + 1) * 64 + cg * 16 + lr];
        o[cg] = wmma_f32(pa[kk], b, o[cg]);
      }
    }
    __syncthreads();
  }

#pragma unroll
  for (int r = 0; r < 8; ++r) {
    float inv = 1.0f / ssum[r];
    int orow = qt * 16 + r + 8 * lh;
    if (orow < kN) {
      float* op = attn_out + (size_t)orow * kC + h * kHD;
      op[lr]      = o[0][r] * inv;
      op[16 + lr] = o[1][r] * inv;
      op[32 + lr] = o[2][r] * inv;
      op[48 + lr] = o[3][r] * inv;
    }
  }
}

// ---------------------------------------------------------------------------
// Kernel 4: output projection.  attn (kN x kC) @ W_proj (kC x kC) + b_proj
// -> d_out (kN x kC) == reference tuple flattened.  Same LDS-shared scheme.
// ---------------------------------------------------------------------------
__global__ __launch_bounds__(256) void proj_gemm_kernel(
    const float* __restrict__ X, const float* __restrict__ W,
    const float* __restrict__ bias, float* __restrict__ out) {
  __shared__ float sW[2][16 * 64];
  int wid = threadIdx.x >> 5;
  int lane = threadIdx.x & 31;
  int lr = lane & 15;
  int lh = lane >> 4;
  int nt = blockIdx.y;
  int mt = blockIdx.x * 8 + wid;
  if (mt >= kMT) mt = kMT - 1;

  int row = mt * 16 + lr;
  int rowc = row < kN ? row : kN - 1;
  const float* xrow = X + (size_t)rowc * kC;

  int srow = threadIdx.x >> 4;
  int scol = (threadIdx.x & 15) * 4;
  const float* wsrc = W + (size_t)srow * kC + nt * 64 + scol;

  v8f acc[4];
#pragma unroll
  for (int c = 0; c < 4; ++c) acc[c] = v8f_zero();

  constexpr int NC = kC / 16;
  copy16_to_lds(wsrc, &sW[0][srow * 64 + scol]);
  for (int c = 0; c < NC; ++c) {
    int cur = c & 1;
    if (c + 1 < NC) {
      copy16_to_lds(wsrc + (size_t)(c + 1) * 16 * kC,
                    &sW[cur ^ 1][srow * 64 + scol]);
      WAIT_ASYNC(1);
    } else {
      WAIT_ASYNC(0);
    }
    __syncthreads();
    const float* wl = sW[cur];
#pragma unroll
    for (int ks = 0; ks < 4; ++ks) {
      int d0 = ks * 4 + 2 * lh;
      v2f a = *(const v2f*)(xrow + c * 16 + d0);
#pragma unroll
      for (int cg = 0; cg < 4; ++cg) {
        v2f b;
        b.x = wl[d0 * 64 + cg * 16 + lr];
        b.y = wl[(d0 + 1) * 64 + cg * 16 + lr];
        acc[cg] = wmma_f32(a, b, acc[cg]);
      }
    }
    __syncthreads();
  }

#pragma unroll
  for (int cg = 0; cg < 4; ++cg) {
    int col = nt * 64 + cg * 16 + lr;
    float bv = bias[col];
#pragma unroll
    for (int r = 0; r < 8; ++r) {
      int orow = mt * 16 + r + 8 * lh;
      if (orow < kN) out[(size_t)orow * kC + col] = acc[cg][r] + bv;
    }
  }
}

// ---------------------------------------------------------------------------
// Host launcher
// ---------------------------------------------------------------------------
extern "C" void kernel_launch(void* const* d_in, const int* in_sizes, int n_in,
                              void* d_out, int out_size, void* d_ws,
                              size_t ws_size, hipStream_t stream) {
  const float* qtok  = (const float*)d_in[0];  // (1, 576, 768)
  const float* ttok  = (const float*)d_in[1];  // (1, 4, 577, 768)
  const float* pos2d = (const float*)d_in[2];  // (1, 576, 2)
  const float* poses = (const float*)d_in[3];  // (1, 4, 4, 4)
  const float* Wqkv  = (const float*)d_in[4];  // (768, 2304)
  const float* bqkv  = (const float*)d_in[5];  // (2304,)
  const float* Wproj = (const float*)d_in[6];  // (768, 768)
  const float* bproj = (const float*)d_in[7];  // (768,)
  float* out = (float*)d_out;                  // (2884, 768) flat

  float* qkv  = (float*)d_ws;                  // kN * kC3 floats
  float* attn = qkv + (size_t)kN * kC3;        // kN * kC floats

  {
    dim3 grid((kMT + 7) / 8, kC3 / 64);        // (23, 36)
    qkv_gemm_kernel<<<grid, 256, 0, stream>>>(qtok, ttok, Wqkv, bqkv, qkv);
  }
  {
    int total = 2 * kN * kNH;
    rope_so3_kernel<<<(total + 255) / 256, 256, 0, stream>>>(qkv, pos2d, poses);
  }
  {
    dim3 grid((kMT + 7) / 8, kNH);             // (23, 12)
    attn_kernel<<<grid, 256, 0, stream>>>(qkv, attn);
  }
  {
    dim3 grid((kMT + 7) / 8, kC / 64);         // (23, 12)
    proj_gemm_kernel<<<grid, 256, 0, stream>>>(attn, Wproj, bproj, out);
  }
}